// TCNNHashGridEncoder_36172214567410
// MI455X (gfx1250) — compile-verified
//
#include <hip/hip_runtime.h>
#include <math.h>
#include <string.h>

// ---- HashGrid config (compile-time, mirrors the reference) ----
#define NUM_LEVELS        16
#define LEVEL_DIM         2
#define LOG2_HASHMAP_SIZE 19
#define HASHMAP_SIZE      (1u << LOG2_HASHMAP_SIZE)

struct LevelParams {
    float scale[NUM_LEVELS];   // float32(np.exp2(l*np.log2(1.3819))*16 - 1)
    int   res[NUM_LEVELS];     // int(np.ceil(scale)) + 1
    int   dense[NUM_LEVELS];   // res^3 <= HASHMAP_SIZE
};

// One thread per (point, level). tid = p*16 + l.
// Output layout [N, 32]: this thread owns out[p*32 + l*2 .. +1] == out[tid*2 .. +1],
// so the float2 store stream is perfectly coalesced across lanes (wave32: 32 x b64).
__global__ __launch_bounds__(256)
void hashgrid_enc_kernel(const float* __restrict__ coords,
                         const float* __restrict__ table,
                         float* __restrict__ out,
                         int n_points,
                         LevelParams P)
{
    const int tid = blockIdx.x * blockDim.x + threadIdx.x;
    const int p = tid >> 4;        // point index
    const int l = tid & 15;        // level index
    if (p >= n_points) return;

    // Streaming coords: prefetch ahead (emits global_prefetch_b8 on gfx1250).
    __builtin_prefetch(coords + (size_t)p * 3 + 3072, 0, 0);

    const float cx = coords[(size_t)p * 3 + 0];
    const float cy = coords[(size_t)p * 3 + 1];
    const float cz = coords[(size_t)p * 3 + 2];

    const float scale  = P.scale[l];
    const int   res    = P.res[l];
    const int   dense  = P.dense[l];

    // pos = coords * scale + 0.5 with the reference's two-rounding sequence
    // (block FMA contraction so f32 numerics match XLA exactly).
    const float px = __fadd_rn(__fmul_rn(cx, scale), 0.5f);
    const float py = __fadd_rn(__fmul_rn(cy, scale), 0.5f);
    const float pz = __fadd_rn(__fmul_rn(cz, scale), 0.5f);

    const float gx = floorf(px), gy = floorf(py), gz = floorf(pz);
    const float fx = px - gx, fy = py - gy, fz = pz - gz;

    const int x0 = (int)gx, y0 = (int)gy, z0 = (int)gz;

    // Per-axis weights indexed by corner bit: w[0] = 1-frac, w[1] = frac.
    const float wx[2] = { 1.0f - fx, fx };
    const float wy[2] = { 1.0f - fy, fy };
    const float wz[2] = { 1.0f - fz, fz };

    // Level's feature table (float2 entries). 64 MB total -> L2-resident.
    const float2* __restrict__ tbl =
        (const float2*)table + (size_t)l * HASHMAP_SIZE;

    const int      rm1  = res - 1;
    const uint32_t mask = HASHMAP_SIZE - 1u;

    float acc0 = 0.0f, acc1 = 0.0f;

#pragma unroll
    for (int c = 0; c < 8; ++c) {
        const int ox = c & 1, oy = (c >> 1) & 1, oz = (c >> 2) & 1;

        // trilinear weight, product order matches reference (d0*d1)*d2
        const float w = __fmul_rn(__fmul_rn(wx[ox], wy[oy]), wz[oz]);

        const int ix = x0 + ox, iy = y0 + oy, iz = z0 + oz;

        // dense stride index (with clip, as in reference)
        const int cxi = min(max(ix, 0), rm1);
        const int cyi = min(max(iy, 0), rm1);
        const int czi = min(max(iz, 0), rm1);
        const uint32_t didx = (uint32_t)(cxi + cyi * res + czi * res * res);

        // spatial hash (instant-ngp primes), mod 2^19
        const uint32_t hidx = (((uint32_t)ix * 1u) ^
                               ((uint32_t)iy * 2654435761u) ^
                               ((uint32_t)iz * 805459861u)) & mask;

        const uint32_t idx = dense ? didx : hidx;   // branchless v_cndmask

        const float2 f = tbl[idx];                  // global_load_b64 gather (L2 hit)
        acc0 = fmaf(w, f.x, acc0);
        acc1 = fmaf(w, f.y, acc1);
    }

    // Non-temporal b64 store: 256 MB output stream must not evict the
    // L2-resident table on MI455X (192 MB L2 holds the full 64 MB table).
    float2 r; r.x = acc0; r.y = acc1;
    unsigned long long bits;
    __builtin_memcpy(&bits, &r, 8);
    __builtin_nontemporal_store(bits,
        (unsigned long long*)(out + (size_t)tid * 2));
}

extern "C" void kernel_launch(void* const* d_in, const int* in_sizes, int n_in,
                              void* d_out, int out_size, void* d_ws, size_t ws_size,
                              hipStream_t stream) {
    const float* coords = (const float*)d_in[0];   // [N, 3] f32
    const float* table  = (const float*)d_in[1];   // [16, 2^19, 2] f32
    float*       out    = (float*)d_out;           // [N, 32] f32

    const int n_points = in_sizes[0] / 3;

    // Per-level constants computed in float64 with the exact numpy sequence:
    //   scale = exp2(l * log2(1.3819)) * 16 - 1 ; res = ceil(scale) + 1
    LevelParams P;
    for (int l = 0; l < NUM_LEVELS; ++l) {
        const double s = exp2((double)l * log2(1.3819)) * 16.0 - 1.0;
        const int    r = (int)ceil(s) + 1;
        P.scale[l] = (float)s;
        P.res[l]   = r;
        P.dense[l] = ((long long)r * r * r <= (long long)HASHMAP_SIZE) ? 1 : 0;
    }

    const long long total   = (long long)n_points * NUM_LEVELS;
    const int       threads = 256;                       // 8 wave32 per block
    const int       blocks  = (int)((total + threads - 1) / threads);

    hashgrid_enc_kernel<<<blocks, threads, 0, stream>>>(coords, table, out,
                                                        n_points, P);
}